// Criterion_60318520705345
// MI455X (gfx1250) — compile-verified
//
#include <hip/hip_runtime.h>
#include <hip/hip_bf16.h>

// ---------------------------------------------------------------------------
// MI455X (gfx1250) loss kernel: piecewise-cosine target + MSE-style reduction.
// Bandwidth-bound: 134 MB streamed once => ~5.8us floor at 23.3 TB/s.
// One wave32 per row; breakpoints hoisted to SGPRs via readfirstlane so the
// segment search cost is paid once per row, not per element.
//  - inner loop unrolled x4 so each wave keeps 4 global loads in flight
//  - loads are non-temporal (stream-once data, no reuse)
//  - wave reduction: V_WMMA_F32_16X16X4_F32 (A=ones) does 32->16 on the
//    matrix unit, then 4 single-instruction ds_swizzle XOR steps finish.
// ---------------------------------------------------------------------------

typedef __attribute__((ext_vector_type(2))) float v2f;
typedef __attribute__((ext_vector_type(8))) float v8f;

#define SEQ 1024
#define WAVES_PER_BLOCK 8

// cos(2*pi*turns) via hardware v_cos_f32 (input in turns; OCML native_cos
// multiplies radians by 1/(2*pi) before this builtin).
__device__ __forceinline__ float cos_2pi(float turns) {
#if __has_builtin(__builtin_amdgcn_cosf)
  return __builtin_amdgcn_cosf(turns);
#else
  return __cosf(turns * 6.28318530717958647692f);
#endif
}

// One ds_swizzle_b32 XOR-reduce step (group-of-32 mode: and=0x1f, xor=mask).
template <int PATTERN>
__device__ __forceinline__ float swz_add(float v) {
  int x = __builtin_amdgcn_ds_swizzle(__float_as_int(v), PATTERN);
  return v + __int_as_float(x);
}

// Full-wave (32 lane) sum. EXEC must be all ones when called (converged code).
// Step 1: v_wmma_f32_16x16x4_f32 with A = ones(16x4), B = {v, 0}:
//   D[i][n] = B[0][n] + B[2][n] = v[n] + v[n+16]; every lane L holds
//   pair(L & 15) in d[0] per the C/D layout.
// Step 2: 4 ds_swizzle XOR steps (8,4,2,1) finish the 16-value reduction.
__device__ __forceinline__ float wave_reduce_sum(float v) {
#if __has_builtin(__builtin_amdgcn_wmma_f32_16x16x4_f32)
  v2f a; a[0] = 1.0f; a[1] = 1.0f;     // A = ones (16x4)
  v2f b; b[0] = v;    b[1] = 0.0f;     // B rows K=0,K=2 carry lanes 0-15/16-31
  v8f c = {};
  v8f d = __builtin_amdgcn_wmma_f32_16x16x4_f32(
      /*neg_a=*/false, a, /*neg_b=*/false, b,
      /*c_mod=*/(short)0, c, /*reuse_a=*/false, /*reuse_b=*/false);
  float s = d[0];                      // = v[L&15] + v[(L&15)+16] in every lane
  s = swz_add<0x201f>(s);              // xor 8
  s = swz_add<0x101f>(s);              // xor 4
  s = swz_add<0x081f>(s);              // xor 2
  s = swz_add<0x041f>(s);              // xor 1
  return s;
#else
  for (int off = 16; off > 0; off >>= 1) v += __shfl_xor(v, off, 32);
  return v;
#endif
}

// Accumulate sum((out[p] - target(p))^2) for positions p in [s, e); the
// covering segment starts at `start` (m3-1024 for the wrap piece) and has
// reciprocal length `inv` (turns per position). s,e,start,inv are
// wave-uniform (SGPR) -> scalar loop control; only the tail is predicated.
__device__ __forceinline__ float do_range(const float* __restrict__ rowPtr,
                                          int lane, int s, int e, int start,
                                          float inv, float acc) {
  int n = e - s;
  if (n <= 0) return acc;
  int full = n >> 5;                         // uniform full-wave iterations
  int p = s + lane;
  float arg  = (float)(p - start) * inv;     // position in turns
  const float step  = 32.0f * inv;
  const float step4 = 128.0f * inv;

  float acc1 = 0.0f;                         // second chain to break fmac dep
  int i = 0;
  for (; i + 4 <= full; i += 4) {
    // Issue 4 streaming loads back-to-back -> 4 loads in flight per wave.
    float o0 = __builtin_nontemporal_load(rowPtr + p);
    float o1 = __builtin_nontemporal_load(rowPtr + p + 32);
    float o2 = __builtin_nontemporal_load(rowPtr + p + 64);
    float o3 = __builtin_nontemporal_load(rowPtr + p + 96);
    float a0 = arg;
    float a1 = a0 + step;
    float a2 = a1 + step;
    float a3 = a2 + step;
    float t0 = __builtin_fmaf(cos_2pi(a0), 0.5f, 0.5f);
    float t1 = __builtin_fmaf(cos_2pi(a1), 0.5f, 0.5f);
    float t2 = __builtin_fmaf(cos_2pi(a2), 0.5f, 0.5f);
    float t3 = __builtin_fmaf(cos_2pi(a3), 0.5f, 0.5f);
    float d0 = o0 - t0;
    float d1 = o1 - t1;
    float d2 = o2 - t2;
    float d3 = o3 - t3;
    acc  = __builtin_fmaf(d0, d0, acc);
    acc1 = __builtin_fmaf(d1, d1, acc1);
    acc  = __builtin_fmaf(d2, d2, acc);
    acc1 = __builtin_fmaf(d3, d3, acc1);
    p += 128;
    arg += step4;
  }
  for (; i < full; ++i) {
    float o  = __builtin_nontemporal_load(rowPtr + p);
    float t  = __builtin_fmaf(cos_2pi(arg), 0.5f, 0.5f);
    float dd = o - t;
    acc = __builtin_fmaf(dd, dd, acc);
    p += 32;
    arg += step;
  }
  if (p < e) {                               // partial tail (diverges, reconverges)
    float o  = __builtin_nontemporal_load(rowPtr + p);
    float t  = __builtin_fmaf(cos_2pi(arg), 0.5f, 0.5f);
    float dd = o - t;
    acc = __builtin_fmaf(dd, dd, acc);
  }
  return acc + acc1;
}

__global__ void __launch_bounds__(256)
criterion_partial_kernel(const float* __restrict__ outputs,
                         const int* __restrict__ indexes,
                         float* __restrict__ partials, int B) {
  const int lane = threadIdx.x & 31;
  const int wave = threadIdx.x >> 5;
  const int row  = blockIdx.x * WAVES_PER_BLOCK + wave;

  float acc = 0.0f;
  if (row < B) {                             // wave-uniform branch, EXEC stays full
    // All lanes load the same 16B of breakpoints (broadcast in L1), then
    // force them into SGPRs so all range bounds are scalar.
    const int4 mi = *(const int4*)(indexes + (size_t)row * 4);
    int m0 = __builtin_amdgcn_readfirstlane(mi.x);
    int m1 = __builtin_amdgcn_readfirstlane(mi.y);
    int m2 = __builtin_amdgcn_readfirstlane(mi.z);
    int m3 = __builtin_amdgcn_readfirstlane(mi.w);

    // 4-element sort network (reference sorts; SALU min/max, once per row).
    int t;
    if (m0 > m1) { t = m0; m0 = m1; m1 = t; }
    if (m2 > m3) { t = m2; m2 = m3; m3 = t; }
    if (m0 > m2) { t = m0; m0 = m2; m2 = t; }
    if (m1 > m3) { t = m1; m1 = m3; m3 = t; }
    if (m1 > m2) { t = m1; m1 = m2; m2 = t; }

    const int l0 = m1 - m0;
    const int l1 = m2 - m1;
    const int l2 = m3 - m2;
    const int l3 = SEQ - m3 + m0;            // circular wrap segment
    const float i0 = 1.0f / (float)l0;
    const float i1 = 1.0f / (float)l1;
    const float i2 = 1.0f / (float)l2;
    const float i3 = 1.0f / (float)l3;

    const float* rowPtr = outputs + (size_t)row * SEQ;

    acc = do_range(rowPtr, lane, m0,  m1,  m0,       i0, acc);
    acc = do_range(rowPtr, lane, m1,  m2,  m1,       i1, acc);
    acc = do_range(rowPtr, lane, m2,  m3,  m2,       i2, acc);
    acc = do_range(rowPtr, lane, m3,  SEQ, m3,       i3, acc);  // wrap, high part
    acc = do_range(rowPtr, lane, 0,   m0,  m3 - SEQ, i3, acc);  // wrap, low part
  }

  // Block reduction: wave sums via WMMA + ds_swizzle, then 8 waves via LDS.
  float ws = wave_reduce_sum(acc);
  __shared__ float smem[WAVES_PER_BLOCK];
  if (lane == 0) smem[wave] = ws;
  __syncthreads();
  if (wave == 0) {
    float bs = (lane < WAVES_PER_BLOCK) ? smem[lane] : 0.0f;
    bs = swz_add<0x101f>(bs);                // xor 4
    bs = swz_add<0x081f>(bs);                // xor 2
    bs = swz_add<0x041f>(bs);                // xor 1
    if (lane == 0) partials[blockIdx.x] = bs;
  }
}

__global__ void __launch_bounds__(256)
criterion_finish_kernel(const float* __restrict__ partials, int P,
                        float* __restrict__ out, float scale) {
  float acc = 0.0f;
  for (int i = threadIdx.x; i < P; i += 256) acc += partials[i];

  const int lane = threadIdx.x & 31;
  const int wave = threadIdx.x >> 5;
  float ws = wave_reduce_sum(acc);
  __shared__ float smem[8];
  if (lane == 0) smem[wave] = ws;
  __syncthreads();
  if (wave == 0) {
    float bs = (lane < 8) ? smem[lane] : 0.0f;
    bs = swz_add<0x101f>(bs);
    bs = swz_add<0x081f>(bs);
    bs = swz_add<0x041f>(bs);
    if (lane == 0) out[0] = bs * scale;
  }
}

extern "C" void kernel_launch(void* const* d_in, const int* in_sizes, int n_in,
                              void* d_out, int out_size, void* d_ws, size_t ws_size,
                              hipStream_t stream) {
  const float* outputs = (const float*)d_in[0];   // [B, 1, SEQ] f32
  const int*   indexes = (const int*)d_in[1];     // [B, 4] i32

  const int B = in_sizes[0] / SEQ;
  const int blocks = (B + WAVES_PER_BLOCK - 1) / WAVES_PER_BLOCK;

  float* partials = (float*)d_ws;                 // blocks * 4 bytes of scratch

  criterion_partial_kernel<<<blocks, 256, 0, stream>>>(outputs, indexes,
                                                       partials, B);

  const float scale = 1.0f / ((float)B * (float)SEQ);
  criterion_finish_kernel<<<1, 256, 0, stream>>>(partials, blocks,
                                                 (float*)d_out, scale);
}